// FuzzyLayer_27702539059254
// MI455X (gfx1250) — compile-verified
//
#include <hip/hip_runtime.h>
#include <hip/hip_bf16.h>

typedef __attribute__((ext_vector_type(2))) float v2f;
typedef __attribute__((ext_vector_type(8))) float v8f;

#define EPS_TRAP 1e-8f
#define EPS_F32  1.1920928955078125e-07f   // np.finfo(np.float32).eps

#define BATCH   4096
#define IN_DIM  64
#define OUT_DIM 64
#define NRULES  256
#define KDIM    768          // 3 powers of w * 256 rules
#define NCOLS   128          // 64 num cols + 64 den cols

// ---------------------------------------------------------------------------
// Kernel 1: build B-matrix (KDIM x NCOLS) of polynomial coefficients from Cons
//   k = r + 256*p  (p = power-1), n in [0,64) -> num coeff, [64,128) -> den
// ---------------------------------------------------------------------------
__global__ __launch_bounds__(256) void fuzzy_coeff(const float* __restrict__ Cons,
                                                   float* __restrict__ Bc) {
    int idx = blockIdx.x * 256 + threadIdx.x;       // 0 .. 98303
    if (idx >= KDIM * NCOLS) return;
    int k = idx >> 7;            // 0..767
    int n = idx & 127;           // 0..127
    int r = k & 255;             // rule
    int p = k >> 8;              // 0,1,2  (w^1, w^2, w^3)
    int o = n & 63;              // output dim
    const float4 q = ((const float4*)Cons)[r * OUT_DIM + o];
    float A = q.x, B = q.y, C = q.z, D = q.w;
    float val;
    if (n < 64) {                // numerator coefficients
        if (p == 0)      val = 3.0f * (D * D - A * A);
        else if (p == 1) val = 3.0f * (C * D - A * B) - 3.0f * (D * D - A * A);
        else             val = (C - D + A - B) * (C - D - A + B);
    } else {                     // denominator coefficients
        if (p == 0)      val = 2.0f * (D - A);
        else if (p == 1) val = (C + A - D - B);
        else             val = 0.0f;
    }
    Bc[idx] = val;
}

// ---------------------------------------------------------------------------
// Kernel 2: rule firing strengths  W[n][r] = min_i trapmf(x[n,i], Antes[r,i,:])
//   16 batch x 16 rule tile per 256-thread block; LDS staged, padded stride 65
// ---------------------------------------------------------------------------
__global__ __launch_bounds__(256) void fuzzy_w(const float* __restrict__ x,
                                               const float* __restrict__ Antes,
                                               float* __restrict__ W) {
    __shared__ float xs[16][65];
    __shared__ float as_[16][65], bs_[16][65], cs_[16][65], ds_[16][65];

    const int tid = threadIdx.x;
    const int bn  = blockIdx.x;      // batch tile  (256 tiles)
    const int br  = blockIdx.y;      // rule tile   (16 tiles)

    // stage x tile: 16 x 64 floats
    for (int idx = tid; idx < 16 * 64; idx += 256) {
        int n = idx >> 6, i = idx & 63;
        xs[n][i] = x[(bn * 16 + n) * IN_DIM + i];
    }
    // stage Antes tile: 16 rules x 64 dims x float4 -> 4 LDS planes
    for (int idx = tid; idx < 16 * 64; idx += 256) {
        int r = idx >> 6, i = idx & 63;
        const float4 q = ((const float4*)Antes)[(br * 16 + r) * IN_DIM + i];
        as_[r][i] = q.x; bs_[r][i] = q.y; cs_[r][i] = q.z; ds_[r][i] = q.w;
    }
    __syncthreads();

    const int n = tid & 15;
    const int r = tid >> 4;
    float wmin = 3.402823466e+38f;
    #pragma unroll 4
    for (int i = 0; i < IN_DIM; ++i) {
        float xv = xs[n][i];
        float a = as_[r][i], b = bs_[r][i], c = cs_[r][i], d = ds_[r][i];
        float rise = (xv - a) / (b - a + EPS_TRAP);
        float fall = (d - xv) / (d - c + EPS_TRAP);
        float y = 0.0f;
        y = (a < xv && xv < b)   ? rise : y;
        y = (b <= xv && xv <= c) ? 1.0f : y;
        y = (c < xv && xv < d)   ? fall : y;
        wmin = fminf(wmin, y);
    }
    W[(bn * 16 + n) * NRULES + (br * 16 + r)] = wmin;
}

// ---------------------------------------------------------------------------
// Kernel 3: WMMA f32 GEMM  [W|W^2|W^3](4096x768) x Bc(768x128) + epilogue
//   One wave32 per 16-row M-tile owns all 128 output columns:
//   8 accumulators (4 num tiles + 4 den tiles). Per K-block of 4:
//     - load lane's w pair ONCE, build w/w^2/w^3 A-fragments in registers
//     - 24 WMMAs (3 powers x 8 tiles), tile-innermost so consecutive WMMAs
//       target different accumulators (8-deep independent chains)
//   W is read exactly once in total (no cross-block / cross-power re-reads).
//   A frag (16x4 f32): lane = M + 16*(K/2), vgpr = K%2  -> 2 floats per lane.
// ---------------------------------------------------------------------------
__global__ __launch_bounds__(32) void fuzzy_gemm(const float* __restrict__ W,
                                                 const float* __restrict__ Bc,
                                                 float* __restrict__ out) {
    const int mtile = blockIdx.x;                 // 0..255
    const int lane  = threadIdx.x;                // 0..31
    const int mrow  = mtile * 16 + (lane & 15);   // A row served by this lane
    const int khalf = (lane >> 4) * 2;            // K sub-offset {0,2}
    const int col   = lane & 15;                  // B col-within-tile this lane serves

    v8f acc[8];                                   // tile t covers cols [t*16, t*16+16)
    #pragma unroll
    for (int t = 0; t < 8; ++t)
        acc[t] = (v8f){0.f, 0.f, 0.f, 0.f, 0.f, 0.f, 0.f, 0.f};

    const float* wrow = W + mrow * NRULES;

    for (int kb = 0; kb < 64; ++kb) {             // 64 K-blocks of 4 per power
        const int r0 = kb * 4 + khalf;            // rule index (power-independent)
        const float w0 = wrow[r0];
        const float w1 = wrow[r0 + 1];
        v2f a[3];
        a[0].x = w0;          a[0].y = w1;          // w
        a[1].x = w0 * w0;     a[1].y = w1 * w1;     // w^2
        a[2].x = a[1].x * w0; a[2].y = a[1].y * w1; // w^3

        #pragma unroll
        for (int p = 0; p < 3; ++p) {
            const int k0 = p * 256 + kb * 4 + khalf;
            const float* brow0 = Bc + (k0    ) * NCOLS + col;
            const float* brow1 = Bc + (k0 + 1) * NCOLS + col;
            v2f b[8];
            #pragma unroll
            for (int t = 0; t < 8; ++t) {
                b[t].x = brow0[t * 16];
                b[t].y = brow1[t * 16];
            }
            #pragma unroll
            for (int t = 0; t < 8; ++t)
                acc[t] = __builtin_amdgcn_wmma_f32_16x16x4_f32(
                             false, a[p], false, b[t], (short)0, acc[t], false, false);
        }
    }

    // Epilogue per 16x16 f32 C/D layout: vgpr j -> M = j + 8*(lane/16), N = lane&15
    // acc[0..3] = num cols 0..63, acc[4..7] = den cols 64..127
    const int orow0 = mtile * 16 + ((lane >> 4) << 3);
    #pragma unroll
    for (int t = 0; t < 4; ++t) {
        #pragma unroll
        for (int j = 0; j < 8; ++j) {
            float den = acc[t + 4][j] + (float)NRULES * EPS_F32;
            float num = acc[t][j];
            float y = (den != 0.0f) ? (num / den) : 0.0f;
            out[(orow0 + j) * OUT_DIM + (t * 16 + col)] = y / 3.0f;
        }
    }
}

// ---------------------------------------------------------------------------
extern "C" void kernel_launch(void* const* d_in, const int* in_sizes, int n_in,
                              void* d_out, int out_size, void* d_ws, size_t ws_size,
                              hipStream_t stream) {
    const float* x     = (const float*)d_in[0];   // 4096 x 64
    const float* Antes = (const float*)d_in[1];   // 256 x 64 x 4
    const float* Cons  = (const float*)d_in[2];   // 256 x 64 x 4
    float* out = (float*)d_out;                   // 4096 x 64

    float* W  = (float*)d_ws;                     // 4096*256 floats (4 MB)
    float* Bc = W + BATCH * NRULES;               // 768*128 floats (384 KB)

    fuzzy_coeff<<<dim3((KDIM * NCOLS + 255) / 256), 256, 0, stream>>>(Cons, Bc);
    fuzzy_w<<<dim3(BATCH / 16, NRULES / 16), 256, 0, stream>>>(x, Antes, W);
    fuzzy_gemm<<<dim3(BATCH / 16), 32, 0, stream>>>(W, Bc, out);
}